// MoE_29927332118881
// MI455X (gfx1250) — compile-verified
//
#include <hip/hip_runtime.h>

typedef __attribute__((ext_vector_type(2))) float v2f;
typedef __attribute__((ext_vector_type(4))) float v4f;
typedef __attribute__((ext_vector_type(8))) float v8f;

#define N_IMG   48            // B*CH = 16*3
#define HW      147456        // 384*384
#define WIDTH   384
#define NCOL    192           // N_IMG * 4 gaussians ("slots")
#define NTILE   12            // 192 / 16 slot tiles
#define TPR     24            // 16-pixel tiles per image row (384/16)
#define INV383  (1.0f / 383.0f)
#define LOG2E   1.4426950408889634f

// d_ws float layout:
//   [0    .. 1536)  A-operand coefficients (pre-scaled by log2(e)), [tile][step][lane][j]
//   [1536 .. 1728)  per-slot mixture weights, [tile][slot]          (12*16)
//
// Slot S = 16*t + m maps to plane n = S>>2, gaussian k = S&3.
// Exponent expanded into polynomial features f = {1, px, py, px^2, py^2, px*py, 0, 0}:
//   log2(e) * (-0.5*q) = sum_f coef[f] * feat[f]   -> exp2() gives exp(-0.5*q)
__global__ __launch_bounds__(NCOL) void gsplat_coef_kernel(const float* __restrict__ params,
                                                           float* __restrict__ ws) {
    const int S = threadIdx.x;                 // slot 0..191
    const int n = S >> 2, k = S & 3;
    const float* p = params + n * 28;          // [mu_x[4], mu_y[4], w[4], Sigma[16]]
    const float mx  = p[k];
    const float my  = p[4 + k];
    const float wgt = p[8 + k];
    const float s00 = p[12 + 4 * k + 0];       // tril: s01 zeroed
    const float s10 = p[12 + 4 * k + 2];
    const float s11 = p[12 + 4 * k + 3];
    // u = s00*d0 + s10*d1 = A1*px + A2*py + A0 ; v = s11*d1 = B2*py + B0
    const float A1 = s00, A2 = s10;
    const float A0 = -(s00 * mx + s10 * my);
    const float B2 = s11, B0 = -s11 * my;
    float cf[8];
    cf[0] = LOG2E * (-0.5f * (A0 * A0 + B0 * B0));   // 1
    cf[1] = LOG2E * (-A1 * A0);                      // px
    cf[2] = LOG2E * (-(A2 * A0 + B2 * B0));          // py
    cf[3] = LOG2E * (-0.5f * A1 * A1);               // px^2
    cf[4] = LOG2E * (-0.5f * (A2 * A2 + B2 * B2));   // py^2
    cf[5] = LOG2E * (-A1 * A2);                      // px*py
    cf[6] = 0.0f;
    cf[7] = 0.0f;
    const int t = S >> 4, m = S & 15;
    // Scatter into WMMA A-operand register layout (16x4, M x K):
    // vgpr j of lane l holds A[m = l&15][k = j + 2*(l>>4)]; feature f = 4*step + 2*hi + j
#pragma unroll
    for (int f = 0; f < 8; ++f) {
        const int s   = f >> 2;
        const int rem = f & 3;
        const int j   = rem & 1;
        const int hi  = rem >> 1;
        const int lane = m + 16 * hi;
        ws[t * 128 + s * 64 + lane * 2 + j] = cf[f];
    }
    ws[1536 + t * 16 + m] = wgt;
}

__global__ __launch_bounds__(256) void gsplat_wmma_kernel(const float* __restrict__ ws,
                                                          float* __restrict__ out) {
    const int tid       = blockIdx.x * 256 + threadIdx.x;
    const int pixTile   = tid >> 5;            // one 16-pixel tile per wave (9216 waves exact)
    const int lane      = threadIdx.x & 31;
    const int c         = lane & 15;           // pixel within tile (D column)
    const int hi        = lane >> 4;
    const int pixelBase = pixTile << 4;
    const int row       = pixTile / TPR;       // output row h -> px
    const int w0        = (pixTile % TPR) << 4;
    const float px = (float)row * INV383;
    const float py = (float)(w0 + c) * INV383;

    // B operand (4x16, K x N): vgpr j of lane l holds B[k = j+2*hi][n = l&15]
    // step0: {1, px | py, px^2} ; step1: {py^2, px*py | 0, 0}
    v2f f0, f1;
    f0[0] = hi ? py      : 1.0f;
    f0[1] = hi ? px * px : px;
    f1[0] = hi ? 0.0f    : py * py;
    f1[1] = hi ? 0.0f    : px * py;

    const float* wsW = ws + 1536;
    // lane's D rows are slots {8*hi .. 8*hi+7} = planes (4t+2hi) and (4t+2hi+1)
    float* o = out + (2 * hi) * HW + pixelBase + c;

    for (int t = 0; t < NTILE; ++t) {
        const float* Ac = ws + t * 128;
        const v2f cf0 = *(const v2f*)(Ac + lane * 2);        // step 0 coeffs (A operand)
        const v2f cf1 = *(const v2f*)(Ac + 64 + lane * 2);   // step 1 coeffs
        const v4f wA  = *(const v4f*)(wsW + t * 16 + 8 * hi);
        const v4f wB  = *(const v4f*)(wsW + t * 16 + 8 * hi + 4);

        v8f acc = {};
        acc = __builtin_amdgcn_wmma_f32_16x16x4_f32(false, cf0, false, f0,
                                                    (short)0, acc, false, false);
        acc = __builtin_amdgcn_wmma_f32_16x16x4_f32(false, cf1, false, f1,
                                                    (short)0, acc, false, false);

        // acc[j] = log2e * (-0.5*q) for slot m = j + 8*hi, pixel c
        const float e0 = __builtin_amdgcn_exp2f(acc[0]);
        const float e1 = __builtin_amdgcn_exp2f(acc[1]);
        const float e2 = __builtin_amdgcn_exp2f(acc[2]);
        const float e3 = __builtin_amdgcn_exp2f(acc[3]);
        const float e4 = __builtin_amdgcn_exp2f(acc[4]);
        const float e5 = __builtin_amdgcn_exp2f(acc[5]);
        const float e6 = __builtin_amdgcn_exp2f(acc[6]);
        const float e7 = __builtin_amdgcn_exp2f(acc[7]);

        // plane A = slots {0..3}+8hi, plane B = slots {4..7}+8hi : intra-lane reduction
        const float gA = (e0 + e1) + (e2 + e3);
        const float gB = (e4 + e5) + (e6 + e7);
        float sA = e0 * wA[0];
        sA = fmaf(e1, wA[1], sA);
        sA = fmaf(e2, wA[2], sA);
        sA = fmaf(e3, wA[3], sA);
        float sB = e4 * wB[0];
        sB = fmaf(e5, wB[1], sB);
        sB = fmaf(e6, wB[2], sB);
        sB = fmaf(e7, wB[3], sB);

        float yA = sA * __builtin_amdgcn_rcpf(fmaxf(1e-7f, gA));
        float yB = sB * __builtin_amdgcn_rcpf(fmaxf(1e-7f, gB));
        yA = fminf(fmaxf(yA, 0.0f), 1.0f);
        yB = fminf(fmaxf(yB, 0.0f), 1.0f);

        o[0]  = yA;   // plane 4t + 2hi
        o[HW] = yB;   // plane 4t + 2hi + 1
        o += 4 * HW;  // next slot tile = +4 planes
    }
}

extern "C" void kernel_launch(void* const* d_in, const int* in_sizes, int n_in,
                              void* d_out, int out_size, void* d_ws, size_t ws_size,
                              hipStream_t stream) {
    (void)in_sizes; (void)n_in; (void)out_size; (void)ws_size;
    // inputs: d_in[0]=height(=384), d_in[1]=width(=384), d_in[2]=params fp32 [16,3,28]
    const float* params = (const float*)d_in[2];
    float* out = (float*)d_out;
    float* ws  = (float*)d_ws;   // needs 1728 floats (6.75 KB)

    gsplat_coef_kernel<<<1, NCOL, 0, stream>>>(params, ws);
    // 384*384/16 = 9216 pixel-tile waves; 8 waves/block -> 1152 blocks
    gsplat_wmma_kernel<<<1152, 256, 0, stream>>>(ws, out);
}